// _BaseFrameEncoder_89833535963167
// MI455X (gfx1250) — compile-verified
//
#include <hip/hip_runtime.h>
#include <hip/hip_bf16.h>

typedef __attribute__((ext_vector_type(2))) float v2f;
typedef __attribute__((ext_vector_type(8))) float v8f;

#define BT      4096   // B*T = 16*256
#define NTOK    55
#define DMODEL  256
#define KTOT    252    // sum of per-MLP padded K
#define NMLP    14
#define NEMB    41

struct EmbTable {
    const int*   idx[NEMB];   // (B,T) int32 index arrays, in token order
    const float* emb[NEMB];   // (V, 256) f32 embedding tables
};

struct MlpDesc {
    const void* src[8];       // concat sources (f32 or int32 per int_mask)
    int         len[8];       // per-source feature length
    int nsrc, din, kpad, xoff, woff, tok;
    unsigned int int_mask;    // bit s set -> src[s] is int32 (cast to f32)
    const float* ln_g; const float* ln_b;
    const float* W;    const float* bias;
};
struct MlpAll { MlpDesc d[NMLP]; };

// ---------------- embedding gather: one block per (pos, token) ----------------
__global__ void emb_gather_kernel(EmbTable tbl, float* __restrict__ out) {
    const int pos = blockIdx.x;      // 0..4095
    const int k   = blockIdx.y;      // 0..40 == token slot
    const int t   = threadIdx.x;     // 0..63, each moves a float4
    const int row = tbl.idx[k][pos];
    const float4* __restrict__ s = (const float4*)(tbl.emb[k] + (size_t)row * DMODEL);
    float4* __restrict__ dp = (float4*)(out + ((size_t)pos * NTOK + k) * DMODEL);
    dp[t] = s[t];
}

// -------- helper: read j-th concat element of MLP m at position pos ----------
__device__ __forceinline__ float read_feat(const MlpDesc& md, int pos, int j) {
    int s = 0;
    while (j >= md.len[s]) { j -= md.len[s]; ++s; }
    if (md.int_mask & (1u << s))
        return (float)(((const int*)md.src[s])[(size_t)pos * md.len[s] + j]);
    return ((const float*)md.src[s])[(size_t)pos * md.len[s] + j];
}

// ---------------- LN prep: pack normalized features into X (BT x KTOT) --------
// Streaming two-pass LayerNorm: no local arrays -> no scratch spills.
__global__ void mlp_prep_kernel(MlpAll all, float* __restrict__ X) {
    const int m   = blockIdx.y;
    const int pos = blockIdx.x * blockDim.x + threadIdx.x;
    if (pos >= BT) return;
    const MlpDesc& md = all.d[m];
    const int din = md.din;

    float sum = 0.f, ss = 0.f;
    for (int j = 0; j < din; ++j) {
        const float v = read_feat(md, pos, j);
        sum += v; ss += v * v;
    }
    const float mu  = sum / (float)din;
    const float var = ss / (float)din - mu * mu;
    const float rs  = rsqrtf(var + 1e-5f);

    float* xp = X + (size_t)pos * KTOT + md.xoff;
    for (int j = 0; j < din; ++j)
        xp[j] = (read_feat(md, pos, j) - mu) * rs * md.ln_g[j] + md.ln_b[j];
    for (int j = din; j < md.kpad; ++j) xp[j] = 0.f;   // zero K padding
}

// ------- pack W into WMMA B-fragment order, zero-padded to kpad --------------
// layout: Wpack[woff + ((c*2 + hi)*DMODEL + col)*2 + j] = W[(4c + 2hi + j)*DMODEL + col]
__global__ void wpack_kernel(MlpAll all, float* __restrict__ Wpack) {
    const int m   = blockIdx.y;
    const int col = blockIdx.x * blockDim.x + threadIdx.x;   // 0..255
    const MlpDesc& md = all.d[m];
    float* wp = Wpack + md.woff;
    for (int k = 0; k < md.kpad; ++k) {
        const float v = (k < md.din) ? md.W[(size_t)k * DMODEL + col] : 0.f;
        const int c  = k >> 2;
        const int hi = (k >> 1) & 1;
        const int j  = k & 1;
        wp[(size_t)((c * 2 + hi) * DMODEL + col) * 2 + j] = v;
    }
}

// ---------------- WMMA GEMM + bias + exact GELU ------------------------------
__device__ __forceinline__ float gelu_exact(float x) {
    return 0.5f * x * (1.0f + erff(x * 0.70710678118654752f));
}

__global__ void mlp_wmma_kernel(MlpAll all, const float* __restrict__ X,
                                const float* __restrict__ Wpack,
                                float* __restrict__ out) {
    const MlpDesc& md = all.d[blockIdx.z];
    const int lane  = threadIdx.x;      // 0..31 (wave32)
    const int laneM = lane & 15;
    const int hi    = lane >> 4;        // 0: K 0/1 half, 1: K 2/3 half
    const int row0  = blockIdx.x * 16;  // 16 positions
    const int col0  = blockIdx.y * 16;  // 16 output channels
    const int col   = col0 + laneM;

    // A: lane = M row, contiguous K pair selected by lane half
    const v2f* __restrict__ ap =
        (const v2f*)(X + (size_t)(row0 + laneM) * KTOT + md.xoff + 2 * hi);
    // B: pre-packed fragment, one aligned b64 per chunk
    const v2f* __restrict__ bp =
        (const v2f*)(Wpack + md.woff + (size_t)(hi * DMODEL + col) * 2);
    const int nchunk = md.kpad >> 2;

    v8f acc = {};
    for (int c = 0; c < nchunk; ++c) {
        const v2f a = ap[c * 2];             // X stride 4 floats per chunk
        const v2f b = bp[c * (DMODEL * 2)];  // Wpack stride 1024 floats per chunk
        acc = __builtin_amdgcn_wmma_f32_16x16x4_f32(
            /*neg_a=*/false, a, /*neg_b=*/false, b,
            /*c_mod=*/(short)0, acc, /*reuse_a=*/false, /*reuse_b=*/false);
    }

    const float bias = md.bias[col];
    const int tok = md.tok;
#pragma unroll
    for (int v = 0; v < 8; ++v) {
        const int r = row0 + v + 8 * hi;   // C/D layout: VGPR v, lane-half picks M / M+8
        const float val = gelu_exact(acc[v] + bias);
        out[((size_t)r * NTOK + tok) * DMODEL + col] = val;
    }
}

// -----------------------------------------------------------------------------
extern "C" void kernel_launch(void* const* d_in, const int* in_sizes, int n_in,
                              void* d_out, int out_size, void* d_ws, size_t ws_size,
                              hipStream_t stream) {
    (void)in_sizes; (void)n_in; (void)out_size; (void)ws_size;
    auto F = [&](int i) { return (const float*)d_in[i]; };
    auto I = [&](int i) { return (const int*)d_in[i]; };

    // ---- embedding tables (input indices per setup_inputs order) ----
    const float* e_stage = F(70); const float* e_port = F(71);
    const float* e_char  = F(72); const float* e_act  = F(73);
    const float* e_cost  = F(74); const float* e_cdir = F(75);
    const float* e_ptype = F(76); const float* e_psub = F(77);

    EmbTable tbl;
    const float* embmap[17] = { e_stage,
        e_port, e_port, e_char, e_char, e_act, e_act, e_cost, e_cost,
        e_cdir, e_cdir, e_char, e_char, e_act, e_act, e_cdir, e_cdir };
    for (int k = 0; k < 17; ++k) { tbl.idx[k] = I(k); tbl.emb[k] = embmap[k]; }
    for (int j = 0; j < 8; ++j) {
        tbl.idx[17 + 3*j] = I(17 + 3*j); tbl.emb[17 + 3*j] = e_port;
        tbl.idx[18 + 3*j] = I(18 + 3*j); tbl.emb[18 + 3*j] = e_ptype;
        tbl.idx[19 + 3*j] = I(19 + 3*j); tbl.emb[19 + 3*j] = e_psub;
    }

    // ---- MLP descriptors ----
    MlpAll all{};
    int xoff = 0;
    auto setm = [&](int slot, int tok, int din, int kpad, int pbase,
                    int nsrc, const int* srcIdx, const int* lens, unsigned imask) {
        MlpDesc& m = all.d[slot];
        for (int s = 0; s < nsrc; ++s) { m.src[s] = d_in[srcIdx[s]]; m.len[s] = lens[s]; }
        m.nsrc = nsrc; m.din = din; m.kpad = kpad;
        m.xoff = xoff; m.woff = xoff * DMODEL;   // Wpack floats consumed so far
        xoff += kpad;
        m.tok = tok; m.int_mask = imask;
        m.ln_g = F(pbase); m.ln_b = F(pbase + 1); m.W = F(pbase + 2); m.bias = F(pbase + 3);
    };

    { int s[1] = {41};        int l[1] = {20};        setm(0,  41, 20, 20,  78, 1, s, l, 0u); }  // glob
    { int s[2] = {42, 58};    int l[2] = {22, 1};     setm(1,  42, 23, 24,  82, 2, s, l, 2u); }  // player self
    { int s[2] = {43, 59};    int l[2] = {22, 1};     setm(2,  43, 23, 24,  82, 2, s, l, 2u); }  // player opp
    { int s[2] = {44, 60};    int l[2] = {27, 1};     setm(3,  44, 28, 28,  86, 2, s, l, 2u); }  // nana self
    { int s[2] = {45, 61};    int l[2] = {27, 1};     setm(4,  45, 28, 28,  86, 2, s, l, 2u); }  // nana opp
    { int s[1] = {46};        int l[1] = {4};         setm(5,  46,  4,  4,  90, 1, s, l, 0u); }  // analog x4
    { int s[1] = {47};        int l[1] = {4};         setm(6,  47,  4,  4,  90, 1, s, l, 0u); }
    { int s[1] = {48};        int l[1] = {4};         setm(7,  48,  4,  4,  90, 1, s, l, 0u); }
    { int s[1] = {49};        int l[1] = {4};         setm(8,  49,  4,  4,  90, 1, s, l, 0u); }
    { int s[8] = {50,51,52,53,54,55,56,57};
      int l[8] = {5,5,5,5,5,5,5,5};                   setm(9,  50, 40, 40,  94, 8, s, l, 0u); }  // projnum
    { int s[2] = {62, 63};    int l[2] = {12, 12};    setm(10, 51, 24, 24,  98, 2, s, l, 3u); }  // btn
    { int s[2] = {64, 65};    int l[2] = {5, 5};      setm(11, 52, 10, 12, 102, 2, s, l, 3u); }  // flag
    { int s[2] = {66, 67};    int l[2] = {12, 12};    setm(12, 53, 24, 24, 106, 2, s, l, 3u); }  // nanabtn
    { int s[2] = {68, 69};    int l[2] = {6, 6};      setm(13, 54, 12, 12, 110, 2, s, l, 3u); }  // nanaflag
    // xoff is now KTOT (252)

    float* out   = (float*)d_out;
    float* X     = (float*)d_ws;                          // BT*KTOT*4  = 4,128,768 B
    float* Wpack = (float*)((char*)d_ws + (size_t)BT * KTOT * 4);  // +258,048 B (8B aligned)

    // 41 embedding tokens: pure streaming gather
    dim3 gEmb(BT, NEMB);
    emb_gather_kernel<<<gEmb, 64, 0, stream>>>(tbl, out);

    // LayerNorm + pack activations
    dim3 gPrep(BT / 64, NMLP);
    mlp_prep_kernel<<<gPrep, 64, 0, stream>>>(all, X);

    // Pack weights into B-fragment order (zero-padded K)
    dim3 gWp(DMODEL / 64, NMLP);
    wpack_kernel<<<gWp, 64, 0, stream>>>(all, Wpack);

    // WMMA GEMM + bias + GELU into the 14 MLP token slots
    dim3 gW(BT / 16, DMODEL / 16, NMLP);
    mlp_wmma_kernel<<<gW, 32, 0, stream>>>(all, X, Wpack, out);
}